// DiagOpLayer_21663815041423
// MI455X (gfx1250) — compile-verified
//
#include <hip/hip_runtime.h>
#include <hip/hip_bf16.h>
#include <math.h>

#define B_   16
#define N_   262144
#define M_   262143        // n-1
#define NM_  131072        // n_miss (= 2^17)
#define NV_  (NM_ / 4)     // float4 elements per batch row = 32768
#define PB_  64            // reduction blocks per batch
#define TPB_ 256

typedef __attribute__((ext_vector_type(2))) float v2f;
typedef __attribute__((ext_vector_type(8))) float v8f;

// Deterministic sum of 64 values (2 per lane) using V_WMMA_F32_16X16X4_F32.
// With B = ones, D[m][n] = sum_k A[m][k]; summing the 8 D VGPRs in a lane gives
// the half-matrix row-sum total; shfl_xor(16) folds the two halves.
__device__ __forceinline__ float wave_sum64(float x, float y) {
  v2f a; a.x = x; a.y = y;
  v2f one; one.x = 1.0f; one.y = 1.0f;
  v8f c = {};
  c = __builtin_amdgcn_wmma_f32_16x16x4_f32(false, a, false, one,
                                            (short)0, c, false, false);
  float t = c[0] + c[1] + c[2] + c[3] + c[4] + c[5] + c[6] + c[7];
  t += __shfl_xor(t, 16);
  return t;  // uniform across the wave
}

// Block (256 threads = 8 waves) deterministic sum; valid in thread 0.
__device__ __forceinline__ float block_sum(float x) {
  __shared__ float lds[TPB_ / 32];
  float wsum = wave_sum64(x, 0.0f);
  if ((threadIdx.x & 31) == 0) lds[threadIdx.x >> 5] = wsum;
  __syncthreads();
  float r = 0.0f;
  if (threadIdx.x == 0) {
#pragma unroll
    for (int i = 0; i < TPB_ / 32; ++i) r += lds[i];
  }
  return r;
}

// w = clip(softplus(u)+1e-3, 1e-4, 1e4); emit w (output) and w^2 (scratch)
__global__ void k_w(const float* __restrict__ u, float* __restrict__ w_out,
                    float* __restrict__ w2) {
  int i = blockIdx.x * blockDim.x + threadIdx.x;
  if (i >= B_ * M_) return;
  float x = u[i];
  float sp = fmaxf(x, 0.0f) + log1pf(expf(-fabsf(x)));  // stable softplus
  float w = fminf(fmaxf(sp + 1e-3f, 1e-4f), 1e4f);
  w_out[i] = w;
  w2[i] = w * w;
}

// Build tridiagonal coefficients + rhs; init v=0, r=rhs, pA=rhs; partial ||rhs||^2
__global__ void k_setup(const float* __restrict__ xf, const int* __restrict__ idx,
                        const float* __restrict__ w2,
                        float* __restrict__ diag, float* __restrict__ offl,
                        float* __restrict__ offr, float* __restrict__ v,
                        float* __restrict__ rcg, float* __restrict__ p,
                        float* __restrict__ part) {
  int b = blockIdx.y;
  const float* w2b = w2 + (size_t)b * M_;
  const float* xb  = xf + (size_t)b * N_;
  size_t base = (size_t)b * NM_;
  float acc = 0.0f;
  for (int j = blockIdx.x * TPB_ + threadIdx.x; j < NM_; j += PB_ * TPB_) {
    int g = idx[j];
    bool ln = (j > 0) && (idx[j - 1] == g - 1);
    bool rn = (j < NM_ - 1) && (idx[j + 1] == g + 1);
    float wl = (g > 0)      ? w2b[g - 1] : 0.0f;
    float wr = (g < N_ - 1) ? w2b[g]     : 0.0f;
    diag[base + j] = wl + wr;
    offl[base + j] = ln ? wl : 0.0f;
    offr[base + j] = rn ? wr : 0.0f;
    float xl = 0.0f, xr = 0.0f;
    if (g > 0 && !ln)      { float t = xb[g - 1]; xl = (t != t) ? 0.0f : t; }
    if (g < N_ - 1 && !rn) { float t = xb[g + 1]; xr = (t != t) ? 0.0f : t; }
    float rhs = wl * xl + wr * xr;
    v[base + j] = 0.0f;
    rcg[base + j] = rhs;
    p[base + j] = rhs;
    acc += rhs * rhs;
  }
  float bs = block_sum(acc);
  if (threadIdx.x == 0) part[b * PB_ + blockIdx.x] = bs;
}

// rs0 = ||r0||^2 ; also init beta = 0 so the fused K-apply yields p0 = r0
__global__ void k_rs0(const float* __restrict__ part, float* __restrict__ rs,
                      float* __restrict__ beta) {
  int b = blockIdx.x, l = threadIdx.x;
  float t = wave_sum64(part[b * PB_ + l], part[b * PB_ + l + 32]);
  if (l == 0) { rs[b] = t; beta[b] = 0.0f; }
}

// Fused: p_new = rcg + beta*p_old ; Kp = K(p_new) ; partial dot(p_new, Kp)
__global__ void k_KpP(const float* __restrict__ diag, const float* __restrict__ offl,
                      const float* __restrict__ offr, const float* __restrict__ rcg,
                      const float* __restrict__ pold, const float* __restrict__ beta,
                      float* __restrict__ pnew, float* __restrict__ Kp,
                      float* __restrict__ part) {
  int b = blockIdx.y;
  float bt = beta[b];
  size_t base = (size_t)b * NM_;
  const float*  rb = rcg + base;
  const float*  qb = pold + base;
  const float4* r4 = (const float4*)rb;
  const float4* q4 = (const float4*)qb;
  const float4* D4 = (const float4*)(diag + base);
  const float4* L4 = (const float4*)(offl + base);
  const float4* R4 = (const float4*)(offr + base);
  float4* P4 = (float4*)(pnew + base);
  float4* K4 = (float4*)(Kp + base);
  float acc = 0.0f;
  for (int vj = blockIdx.x * TPB_ + threadIdx.x; vj < NV_; vj += PB_ * TPB_) {
    float4 rv = r4[vj];
    float4 qv = q4[vj];
    float4 pn;
    pn.x = rv.x + bt * qv.x;
    pn.y = rv.y + bt * qv.y;
    pn.z = rv.z + bt * qv.z;
    pn.w = rv.w + bt * qv.w;
    float pnl = 0.0f, pnr = 0.0f;
    if (vj > 0)        { int jm = 4 * vj - 1; pnl = rb[jm] + bt * qb[jm]; }
    if (vj < NV_ - 1)  { int jp = 4 * vj + 4; pnr = rb[jp] + bt * qb[jp]; }
    float4 D = D4[vj], L = L4[vj], R = R4[vj];
    float4 k;
    k.x = D.x * pn.x - L.x * pnl  - R.x * pn.y;
    k.y = D.y * pn.y - L.y * pn.x - R.y * pn.z;
    k.z = D.z * pn.z - L.z * pn.y - R.z * pn.w;
    k.w = D.w * pn.w - L.w * pn.z - R.w * pnr;
    P4[vj] = pn;
    K4[vj] = k;
    acc += pn.x * k.x + pn.y * k.y + pn.z * k.z + pn.w * k.w;
  }
  float bs = block_sum(acc);
  if (threadIdx.x == 0) part[b * PB_ + blockIdx.x] = bs;
}

__global__ void k_alpha(const float* __restrict__ part, const float* __restrict__ rs,
                        float* __restrict__ alpha) {
  int b = blockIdx.x, l = threadIdx.x;
  float t = wave_sum64(part[b * PB_ + l], part[b * PB_ + l + 32]);
  if (l == 0) alpha[b] = rs[b] / (t + 1e-30f);
}

// v += alpha*p ; r -= alpha*Kp ; partial ||r||^2
__global__ void k_upd(const float* __restrict__ alpha, const float* __restrict__ p,
                      const float* __restrict__ Kp, float* __restrict__ v,
                      float* __restrict__ rcg, float* __restrict__ part) {
  int b = blockIdx.y;
  float al = alpha[b];
  size_t base = (size_t)b * NM_;
  const float4* p4 = (const float4*)(p + base);
  const float4* K4 = (const float4*)(Kp + base);
  float4* v4 = (float4*)(v + base);
  float4* r4 = (float4*)(rcg + base);
  float acc = 0.0f;
  for (int vj = blockIdx.x * TPB_ + threadIdx.x; vj < NV_; vj += PB_ * TPB_) {
    float4 pv = p4[vj], kv = K4[vj], vv = v4[vj], rv = r4[vj];
    vv.x += al * pv.x; vv.y += al * pv.y; vv.z += al * pv.z; vv.w += al * pv.w;
    rv.x -= al * kv.x; rv.y -= al * kv.y; rv.z -= al * kv.z; rv.w -= al * kv.w;
    v4[vj] = vv;
    r4[vj] = rv;
    acc += rv.x * rv.x + rv.y * rv.y + rv.z * rv.z + rv.w * rv.w;
  }
  float bs = block_sum(acc);
  if (threadIdx.x == 0) part[b * PB_ + blockIdx.x] = bs;
}

__global__ void k_beta(const float* __restrict__ part, float* __restrict__ rs,
                       float* __restrict__ beta) {
  int b = blockIdx.x, l = threadIdx.x;
  float t = wave_sum64(part[b * PB_ + l], part[b * PB_ + l + 32]);
  if (l == 0) { beta[b] = t / (rs[b] + 1e-30f); rs[b] = t; }
}

__global__ void k_y1(const float4* __restrict__ xf, float4* __restrict__ y) {
  int i = blockIdx.x * blockDim.x + threadIdx.x;  // i < B_*N_/4 exactly
  float4 t = xf[i];
  float4 o;
  o.x = (t.x != t.x) ? 0.0f : t.x;
  o.y = (t.y != t.y) ? 0.0f : t.y;
  o.z = (t.z != t.z) ? 0.0f : t.z;
  o.w = (t.w != t.w) ? 0.0f : t.w;
  y[i] = o;
}

__global__ void k_y2(const int* __restrict__ idx, const float* __restrict__ v,
                     float* __restrict__ y) {
  int i = blockIdx.x * blockDim.x + threadIdx.x;  // i < B_*NM_ exactly
  int b = i >> 17;
  int j = i & (NM_ - 1);
  y[(size_t)b * N_ + idx[j]] = v[i];
}

// r = D(y) to d_out; partial phi = sum w2*r*r
__global__ void k_final(const float* __restrict__ y, const float* __restrict__ w2,
                        float* __restrict__ r_out, float* __restrict__ part) {
  int b = blockIdx.y;
  const float* yb = y + (size_t)b * N_;
  size_t baseM = (size_t)b * M_;
  float acc = 0.0f;
  const int stride = PB_ * TPB_;
  const int bound = ((M_ + stride - 1) / stride) * stride;  // uniform trip count
  for (int i = blockIdx.x * TPB_ + threadIdx.x; i < bound; i += stride) {
    if (i < M_) {
      float r = yb[i + 1] - yb[i];
      r_out[baseM + i] = r;
      acc += w2[baseM + i] * r * r;
    }
  }
  float bs = block_sum(acc);
  if (threadIdx.x == 0) part[b * PB_ + blockIdx.x] = bs;
}

__global__ void k_phi(const float* __restrict__ part, float* __restrict__ phi) {
  int b = blockIdx.x, l = threadIdx.x;
  float t = wave_sum64(part[b * PB_ + l], part[b * PB_ + l + 32]);
  if (l == 0) phi[b] = t;
}

extern "C" void kernel_launch(void* const* d_in, const int* in_sizes, int n_in,
                              void* d_out, int out_size, void* d_ws, size_t ws_size,
                              hipStream_t stream) {
  (void)in_sizes; (void)n_in; (void)out_size; (void)ws_size;
  const float* u   = (const float*)d_in[0];  // [B, n-1]
  const float* xf  = (const float*)d_in[1];  // [B, n]
  const int*   idx = (const int*)d_in[2];    // [n_miss]

  float* out = (float*)d_out;                // phi | v | r | w
  float* phi = out;
  float* v   = out + B_;
  float* rO  = v + (size_t)B_ * NM_;
  float* wO  = rO + (size_t)B_ * M_;

  float* ws = (float*)d_ws;
  size_t o = 0;
  float* w2   = ws + o; o += (size_t)B_ * M_;   // 16-float multiple -> 64B aligned
  float* diag = ws + o; o += (size_t)B_ * NM_;
  float* offl = ws + o; o += (size_t)B_ * NM_;
  float* offr = ws + o; o += (size_t)B_ * NM_;
  float* rcg  = ws + o; o += (size_t)B_ * NM_;
  float* pA   = ws + o; o += (size_t)B_ * NM_;
  float* pB   = ws + o; o += (size_t)B_ * NM_;  // pA..pB contiguous = B_*N_ floats
  float* Kp   = ws + o; o += (size_t)B_ * NM_;
  float* y    = pA;                             // alias over dead pA/pB after CG
  float* part = ws + o; o += (size_t)B_ * PB_;
  float* rs   = ws + o; o += B_;
  float* alpha= ws + o; o += B_;
  float* beta = ws + o; o += B_;

  dim3 red(PB_, B_);
  k_w<<<(B_ * M_ + TPB_ - 1) / TPB_, TPB_, 0, stream>>>(u, wO, w2);
  k_setup<<<red, TPB_, 0, stream>>>(xf, idx, w2, diag, offl, offr, v, rcg, pA, part);
  k_rs0<<<B_, 32, 0, stream>>>(part, rs, beta);

  for (int it = 0; it < 20; ++it) {
    float* pold = (it & 1) ? pB : pA;
    float* pnew = (it & 1) ? pA : pB;
    k_KpP<<<red, TPB_, 0, stream>>>(diag, offl, offr, rcg, pold, beta, pnew, Kp, part);
    k_alpha<<<B_, 32, 0, stream>>>(part, rs, alpha);
    k_upd<<<red, TPB_, 0, stream>>>(alpha, pnew, Kp, v, rcg, part);
    k_beta<<<B_, 32, 0, stream>>>(part, rs, beta);
  }

  k_y1<<<(B_ * N_ / 4) / TPB_, TPB_, 0, stream>>>((const float4*)xf, (float4*)y);
  k_y2<<<(B_ * NM_) / TPB_, TPB_, 0, stream>>>(idx, v, y);
  k_final<<<red, TPB_, 0, stream>>>(y, w2, rO, part);
  k_phi<<<B_, 32, 0, stream>>>(part, phi);
}